// GNN_31997506355646
// MI455X (gfx1250) — compile-verified
//
#include <hip/hip_runtime.h>

typedef float v2f __attribute__((ext_vector_type(2)));
typedef float v8f __attribute__((ext_vector_type(8)));

// ---------------------------------------------------------------------------
// GEMM H = X * W via V_WMMA_F32_16X16X4_F32 (full fp32 precision).
// X: [M x K] row-major, W: [K x Ncols] row-major, H: [M x Ncols] row-major.
// One wave32 produces one 16x16 tile; K-loop steps 4 per WMMA.
// M % 16 == 0, Ncols % 16 == 0, K % 4 == 0 (holds: 50000, 128/64, 128).
// ---------------------------------------------------------------------------
__global__ __launch_bounds__(256) void gcn_gemm_f32_wmma(
    const float* __restrict__ X, const float* __restrict__ W,
    float* __restrict__ H, int M, int K, int Ncols)
{
    const int wave = (int)((blockIdx.x * blockDim.x + threadIdx.x) >> 5);
    const int lane = (int)(threadIdx.x & 31);
    const int tilesN = Ncols >> 4;
    const int total  = (M >> 4) * tilesN;
    if (wave >= total) return;            // wave-uniform: EXEC stays all-1s

    const int tm = wave / tilesN;
    const int tn = wave - tm * tilesN;
    const int m0 = tm << 4;
    const int n0 = tn << 4;

    const int hi = lane >> 4;             // 0: lanes 0-15, 1: lanes 16-31
    const int l  = lane & 15;

    // A fragment: lane<16 holds (M=l, K=k..k+1); lane>=16 holds (M=l, K=k+2..k+3)
    const float* arow = X + (size_t)(m0 + l) * K + (hi << 1);
    // B fragment: lane<16 VGPR0/1 = rows k,k+1 at col n0+l; lane>=16 rows k+2,k+3
    const float* bptr = W + (size_t)(hi << 1) * Ncols + (n0 + l);

    v8f acc = {};
    for (int k = 0; k < K; k += 4) {
        v2f a, b;
        a.x = arow[k];
        a.y = arow[k + 1];
        b.x = bptr[(size_t)k * Ncols];
        b.y = bptr[(size_t)(k + 1) * Ncols];
        acc = __builtin_amdgcn_wmma_f32_16x16x4_f32(
            /*neg_a=*/false, a, /*neg_b=*/false, b,
            /*c_mod=*/(short)0, acc, /*reuse_a=*/false, /*reuse_b=*/false);
    }

    // D layout: VGPR r -> M = r (+8 for upper lane half), N = l
    float* hp = H + (size_t)(m0 + (hi << 3)) * Ncols + n0 + l;
#pragma unroll
    for (int r = 0; r < 8; ++r)
        hp[(size_t)r * Ncols] = acc[r];
}

// ---------------------------------------------------------------------------
// CSR construction (once; reused by all 3 layers)
// ---------------------------------------------------------------------------
__global__ void gcn_zero_i32(int* __restrict__ p, int n)
{
    int i = blockIdx.x * blockDim.x + threadIdx.x;
    if (i < n) p[i] = 0;
}

__global__ void gcn_count_dst(const int* __restrict__ dst, int* __restrict__ cnt, int e)
{
    int i = blockIdx.x * blockDim.x + threadIdx.x;
    if (i < e) atomicAdd(&cnt[dst[i]], 1);
}

// dis[i] = (1 + in_degree)^(-1/2)   (self-loop included; always > 0)
__global__ void gcn_dis_from_cnt(const int* __restrict__ cnt, float* __restrict__ dis, int n)
{
    int i = blockIdx.x * blockDim.x + threadIdx.x;
    if (i < n) dis[i] = rsqrtf(1.0f + (float)cnt[i]);
}

// Single-workgroup exclusive scan: rowp[0..n] from cnt[0..n-1].
__global__ __launch_bounds__(1024) void gcn_scan_rowptr(
    const int* __restrict__ cnt, int* __restrict__ rowp, int n)
{
    __shared__ int sbuf[1024];
    __shared__ int scarry;
    if (threadIdx.x == 0) scarry = 0;
    __syncthreads();

    for (int base = 0; base < n; base += 1024) {
        int i = base + (int)threadIdx.x;
        int v = (i < n) ? cnt[i] : 0;
        sbuf[threadIdx.x] = v;
        __syncthreads();
        // Hillis-Steele inclusive scan over 1024 elements
        for (int off = 1; off < 1024; off <<= 1) {
            int t = (threadIdx.x >= (unsigned)off) ? sbuf[threadIdx.x - off] : 0;
            __syncthreads();
            sbuf[threadIdx.x] += t;
            __syncthreads();
        }
        int carry = scarry;
        if (i < n) rowp[i] = carry + sbuf[threadIdx.x] - v;   // exclusive
        __syncthreads();
        if (threadIdx.x == 0) scarry = carry + sbuf[1023];
        __syncthreads();
    }
    if (threadIdx.x == 0) rowp[n] = scarry;
}

// Bucket-fill: esrc[rowp[d] + pos] = src(e) for every edge e with dst(e)==d.
__global__ void gcn_fill_csr(const int* __restrict__ src, const int* __restrict__ dst,
                             const int* __restrict__ rowp, int* __restrict__ cur,
                             int* __restrict__ esrc, int e)
{
    int i = blockIdx.x * blockDim.x + threadIdx.x;
    if (i >= e) return;
    int d = dst[i];
    int p = atomicAdd(&cur[d], 1);
    esrc[rowp[d] + p] = src[i];
}

// ---------------------------------------------------------------------------
// Pull-mode aggregation: one wave32 per destination node. No fp32 atomics.
// out[d] = act( sum_{s in N(d)} h[s]*dis[s]*dis[d] + h[d]*dis[d]^2 + bias )
// D = 128: lane handles 4 contiguous floats (b128 loads).
// ---------------------------------------------------------------------------
__global__ __launch_bounds__(256) void gcn_gather_d128(
    const float* __restrict__ H, const int* __restrict__ rowp,
    const int* __restrict__ esrc, const float* __restrict__ dis,
    const float* __restrict__ bias, float* __restrict__ out,
    int N, int relu)
{
    const int wave = (int)((blockIdx.x * blockDim.x + threadIdx.x) >> 5);
    const int lane = (int)(threadIdx.x & 31);
    if (wave >= N) return;

    const int d  = wave;
    const int f0 = lane << 2;                       // 4 floats per lane, D=128
    const float dd = dis[d];

    float4 acc = *(const float4*)(H + (size_t)d * 128 + f0);
    const float sw = dd * dd;                       // self-loop weight
    acc.x *= sw; acc.y *= sw; acc.z *= sw; acc.w *= sw;

    const int beg = rowp[d], end = rowp[d + 1];
    for (int j = beg; j < end; ++j) {
        int s = esrc[j];
        float nrm = dis[s] * dd;
        float4 hv = *(const float4*)(H + (size_t)s * 128 + f0);
        acc.x += hv.x * nrm; acc.y += hv.y * nrm;
        acc.z += hv.z * nrm; acc.w += hv.w * nrm;
    }

    float4 bv = *(const float4*)(bias + f0);
    acc.x += bv.x; acc.y += bv.y; acc.z += bv.z; acc.w += bv.w;
    if (relu) {
        acc.x = fmaxf(acc.x, 0.0f); acc.y = fmaxf(acc.y, 0.0f);
        acc.z = fmaxf(acc.z, 0.0f); acc.w = fmaxf(acc.w, 0.0f);
    }
    *(float4*)(out + (size_t)d * 128 + f0) = acc;
}

// D = 64: lane handles 2 contiguous floats (b64 loads).
__global__ __launch_bounds__(256) void gcn_gather_d64(
    const float* __restrict__ H, const int* __restrict__ rowp,
    const int* __restrict__ esrc, const float* __restrict__ dis,
    const float* __restrict__ bias, float* __restrict__ out,
    int N, int relu)
{
    const int wave = (int)((blockIdx.x * blockDim.x + threadIdx.x) >> 5);
    const int lane = (int)(threadIdx.x & 31);
    if (wave >= N) return;

    const int d  = wave;
    const int f0 = lane << 1;                       // 2 floats per lane, D=64
    const float dd = dis[d];

    float2 acc = *(const float2*)(H + (size_t)d * 64 + f0);
    const float sw = dd * dd;
    acc.x *= sw; acc.y *= sw;

    const int beg = rowp[d], end = rowp[d + 1];
    for (int j = beg; j < end; ++j) {
        int s = esrc[j];
        float nrm = dis[s] * dd;
        float2 hv = *(const float2*)(H + (size_t)s * 64 + f0);
        acc.x += hv.x * nrm; acc.y += hv.y * nrm;
    }

    float2 bv = *(const float2*)(bias + f0);
    acc.x += bv.x; acc.y += bv.y;
    if (relu) { acc.x = fmaxf(acc.x, 0.0f); acc.y = fmaxf(acc.y, 0.0f); }
    *(float2*)(out + (size_t)d * 64 + f0) = acc;
}

// ---------------------------------------------------------------------------
// Host-side orchestration
// ---------------------------------------------------------------------------
static inline int cdiv_ll(long long a, long long b) { return (int)((a + b - 1) / b); }

static void launch_gemm(const float* X, const float* W, float* H,
                        int M, int K, int Ncols, hipStream_t stream)
{
    long long waves = (long long)(M >> 4) * (Ncols >> 4);
    long long threads = waves * 32;
    gcn_gemm_f32_wmma<<<cdiv_ll(threads, 256), 256, 0, stream>>>(X, W, H, M, K, Ncols);
}

extern "C" void kernel_launch(void* const* d_in, const int* in_sizes, int n_in,
                              void* d_out, int out_size, void* d_ws, size_t ws_size,
                              hipStream_t stream)
{
    (void)n_in; (void)out_size; (void)ws_size;

    const float* x  = (const float*)d_in[0];
    const int*   ei = (const int*)d_in[1];
    const float* W1 = (const float*)d_in[2];
    const float* b1 = (const float*)d_in[3];
    const float* W2 = (const float*)d_in[4];
    const float* b2 = (const float*)d_in[5];
    const float* W3 = (const float*)d_in[6];
    const float* b3 = (const float*)d_in[7];
    float* out = (float*)d_out;

    const int D_HID = in_sizes[3];                 // 128
    const int D_IN  = in_sizes[2] / D_HID;         // 128
    const int D_OUT = in_sizes[7];                 // 64
    const int N     = in_sizes[0] / D_IN;          // 50000
    const int E     = in_sizes[1] / 2;             // 800000

    const int* src = ei;
    const int* dst = ei + E;

    // Workspace carve:
    //   cnt[N] | cur[N] | rowp[N+1 (+pad)] | esrc[E] | dis[N] | hbuf[N*128] | act[N*128]
    int*   cnt  = (int*)d_ws;
    int*   cur  = cnt + N;
    int*   rowp = cur + N;
    int*   esrc = rowp + (N + 8);                  // +8 keeps later carve 16B-aligned
    float* dis  = (float*)(esrc + E);
    float* hbuf = dis + N;
    float* act  = hbuf + (size_t)N * D_HID;

    const int nthr = 256;

    // ---- CSR build + normalization (once, reused for all layers) ----
    gcn_zero_i32<<<cdiv_ll(N, nthr), nthr, 0, stream>>>(cnt, N);
    gcn_count_dst<<<cdiv_ll(E, nthr), nthr, 0, stream>>>(dst, cnt, E);
    gcn_dis_from_cnt<<<cdiv_ll(N, nthr), nthr, 0, stream>>>(cnt, dis, N);
    gcn_scan_rowptr<<<1, 1024, 0, stream>>>(cnt, rowp, N);
    gcn_zero_i32<<<cdiv_ll(N, nthr), nthr, 0, stream>>>(cur, N);
    gcn_fill_csr<<<cdiv_ll(E, nthr), nthr, 0, stream>>>(src, dst, rowp, cur, esrc, E);

    const int gwaves = cdiv_ll((long long)N * 32, nthr);

    // ---- Layer 1: x -> act (ReLU) ----
    launch_gemm(x, W1, hbuf, N, D_IN, D_HID, stream);
    gcn_gather_d128<<<gwaves, nthr, 0, stream>>>(hbuf, rowp, esrc, dis, b1, act, N, 1);

    // ---- Layer 2: act -> act (ReLU) ----
    launch_gemm(act, W2, hbuf, N, D_HID, D_HID, stream);
    gcn_gather_d128<<<gwaves, nthr, 0, stream>>>(hbuf, rowp, esrc, dis, b2, act, N, 1);

    // ---- Layer 3: act -> d_out (no ReLU) ----
    launch_gemm(act, W3, hbuf, N, D_HID, D_OUT, stream);
    gcn_gather_d64<<<gwaves, nthr, 0, stream>>>(hbuf, rowp, esrc, dis, b3, out, N, 0);
}